// ourModel_29549374997148
// MI455X (gfx1250) — compile-verified
//
#include <hip/hip_runtime.h>
#include <math.h>

// ---------------- problem constants (from reference) ----------------
#define N_NODES 100000
#define N_EDGES 1600000
#define IN_DIM  128
#define HID     128
#define NCLS    10
#define ENC_H   32
#define ENC_OUT 4
#define BN_EPS  1e-5f

typedef __attribute__((ext_vector_type(2))) float v2f;
typedef __attribute__((ext_vector_type(8))) float v8f;

// =====================================================================
// degree / norm precompute
// =====================================================================
__global__ void init_deg_kernel(float* deg) {
    int i = blockIdx.x * blockDim.x + threadIdx.x;
    if (i < N_NODES) deg[i] = 1.0f;               // self loop
}

__global__ void edge_deg_kernel(const int* __restrict__ ei, float* deg) {
    int e = blockIdx.x * blockDim.x + threadIdx.x;
    if (e < N_EDGES) atomicAdd(&deg[ei[N_EDGES + e]], 1.0f);
}

__global__ void dinv_kernel(const float* __restrict__ deg, float* dinv) {
    int i = blockIdx.x * blockDim.x + threadIdx.x;
    if (i < N_NODES) dinv[i] = rsqrtf(deg[i]);    // deg >= 1 always
}

__global__ void zero_kernel(float* p, int n) {
    int i = blockIdx.x * blockDim.x + threadIdx.x;
    if (i < n) p[i] = 0.0f;
}

// =====================================================================
// GEMM: Y[M x dout] = X[M x 128] * W[128 x dout], f32 WMMA 16x16x4.
// W staged in LDS in K-pair-major float2 layout so each B fragment is a
// single aligned ds_load_b64:
//   wl2[(k>>1)*ldn + n] = { W[k][n], W[k+1][n] }
// Each wave owns one 16-wide N tile; the block sweeps M tiles grid-stride
// to amortize the LDS fill; next M tile's A rows are prefetched
// (global_prefetch_b8) while the current 32-WMMA chain runs.
// =====================================================================
__global__ void gemm_wmma_kernel(const float* __restrict__ X,
                                 const float* __restrict__ W,
                                 float* __restrict__ Y,
                                 int dout, int ldy, int mtiles) {
    __shared__ v2f wl2[64 * 128];                 // 64 KB, max panel
    const int ldn = (dout + 15) & ~15;

    for (int idx = threadIdx.x; idx < 64 * ldn; idx += blockDim.x) {
        int kp = idx / ldn, nn = idx - kp * ldn;
        v2f w = {0.0f, 0.0f};
        if (nn < dout) {
            w.x = W[(2 * kp)     * dout + nn];
            w.y = W[(2 * kp + 1) * dout + nn];
        }
        wl2[idx] = w;
    }
    __syncthreads();

    const int wid    = threadIdx.x >> 5;
    const int lane   = threadIdx.x & 31;
    const int ntiles = ldn >> 4;
    if (wid >= ntiles) return;                    // whole waves retire; EXEC stays all-1s

    const int n0  = wid << 4;
    const int lm  = lane & 15;                    // row (A) / col (B,D) within tile
    const int lkp = lane >> 4;                    // K-pair select within a K-step

    const v2f* bbase = wl2 + lkp * ldn + n0 + lm; // step stride: 2*ldn v2f

    for (int mt = blockIdx.x; mt < mtiles; mt += gridDim.x) {
        const int m0 = mt << 4;
        // prefetch next M tile's A rows into cache while we compute
        int mtn = mt + gridDim.x;
        if (mtn < mtiles) {
            const char* pf = (const char*)(X + (size_t)((mtn << 4) + lm) * 128);
            __builtin_prefetch(pf + lkp * 256, 0, 1);
        }
        const v2f* arow = (const v2f*)(X + (size_t)(m0 + lm) * 128) + lkp;
        v8f acc = {};
        #pragma unroll 8
        for (int ks = 0; ks < 32; ++ks) {         // 32 K-steps of 4
            v2f a = arow[ks * 2];                 // global_load_b64
            v2f b = bbase[ks * 2 * ldn];          // ds_load_b64
            acc = __builtin_amdgcn_wmma_f32_16x16x4_f32(
                false, a, false, b, (short)0, acc, false, false);
        }
        #pragma unroll
        for (int r = 0; r < 8; ++r) {
            int m = m0 + r + (lkp << 3);
            Y[m * ldy + n0 + lm] = acc[r];
        }
    }
}

// =====================================================================
// Encoder: per node n with class c = max(lab,0):
//   h = relu(xl[n] @ eW1[c] + eb1[c])   (dout -> 32)
//   enc[n] = h @ eW2[c] + eb2[c]        (32 -> 4), zeroed if lab == -1
// one wave per node; lane k owns h_k; butterfly-reduce for the 4 outputs.
// =====================================================================
__global__ void encoder_kernel(const float* __restrict__ xl, int dout, int ld,
                               const int* __restrict__ lab,
                               const float* __restrict__ eW1,
                               const float* __restrict__ eb1,
                               const float* __restrict__ eW2,
                               const float* __restrict__ eb2,
                               float* __restrict__ enc) {
    int wid  = (blockIdx.x * blockDim.x + threadIdx.x) >> 5;
    int lane = threadIdx.x & 31;
    if (wid >= N_NODES) return;

    int l = lab[wid];
    int c = l < 0 ? 0 : l;
    const float* xr = xl + (size_t)wid * ld;
    const float* w1 = eW1 + (size_t)c * dout * ENC_H;

    float h = eb1[c * ENC_H + lane];
    for (int d = 0; d < dout; ++d)
        h = fmaf(xr[d], w1[d * ENC_H + lane], h);
    h = fmaxf(h, 0.0f);

    const float* w2 = eW2 + ((size_t)c * ENC_H + lane) * ENC_OUT;
    float p0 = h * w2[0], p1 = h * w2[1], p2 = h * w2[2], p3 = h * w2[3];
    #pragma unroll
    for (int off = 16; off > 0; off >>= 1) {
        p0 += __shfl_xor(p0, off);
        p1 += __shfl_xor(p1, off);
        p2 += __shfl_xor(p2, off);
        p3 += __shfl_xor(p3, off);
    }
    if (lane == 0) {
        float4 e;
        if (l < 0) { e.x = e.y = e.z = e.w = 0.0f; }
        else {
            const float* b2 = eb2 + c * ENC_OUT;
            e.x = p0 + b2[0]; e.y = p1 + b2[1]; e.z = p2 + b2[2]; e.w = p3 + b2[3];
        }
        ((float4*)enc)[wid] = e;
    }
}

// =====================================================================
// Edge scatter: for edge e (incl. self loops), dst class cd:
//   h1 = relu(w_e * (enc[src] @ dW1[cd]) + db1[cd])   (32 lanes = 32 outs)
//   s[dst] += h1   (32 coalesced f32 atomics, L2-resident target)
// dW1 (5 KB) + db1 (1.25 KB) staged in LDS. One wave per edge, grid-stride.
// =====================================================================
__global__ void edge_kernel(const int* __restrict__ ei,
                            const int* __restrict__ lab,
                            const float* __restrict__ dinv,
                            const float* __restrict__ enc,
                            const float* __restrict__ dW1,
                            const float* __restrict__ db1,
                            float* __restrict__ s) {
    __shared__ float w1[NCLS * ENC_OUT * ENC_H];  // [10][4][32]
    __shared__ float b1[NCLS * ENC_H];            // [10][32]
    for (int i = threadIdx.x; i < NCLS * ENC_OUT * ENC_H; i += blockDim.x) w1[i] = dW1[i];
    for (int i = threadIdx.x; i < NCLS * ENC_H; i += blockDim.x) b1[i] = db1[i];
    __syncthreads();

    const int lane = threadIdx.x & 31;
    const int wpb  = blockDim.x >> 5;
    const int total = N_EDGES + N_NODES;
    for (int e = blockIdx.x * wpb + (threadIdx.x >> 5); e < total;
         e += gridDim.x * wpb) {
        int r, cl;
        if (e < N_EDGES) { r = ei[e]; cl = ei[N_EDGES + e]; }
        else             { r = cl = e - N_EDGES; }
        float w  = dinv[r] * dinv[cl];
        int   lc = lab[cl];
        int   c  = lc < 0 ? 0 : lc;
        const float4 ev = ((const float4*)enc)[r];
        const float* wc = w1 + c * (ENC_OUT * ENC_H);
        float t = ev.x * wc[lane]
                + ev.y * wc[ENC_H + lane]
                + ev.z * wc[2 * ENC_H + lane]
                + ev.w * wc[3 * ENC_H + lane];
        float h1 = fmaxf(fmaf(w, t, b1[c * ENC_H + lane]), 0.0f);
        atomicAdd(&s[(size_t)cl * ENC_H + lane], h1);
    }
}

// =====================================================================
// Finalize (in place on xl):
//   mh = s[n]/deg[n];  dec = mh @ dW2[c] + db2[c]
//   out = (lab==-1 ? xl[n] : dec) + bias;  optional BN(eval) + relu
// one wave per node; lane k holds mh_k, broadcast by shuffle.
// =====================================================================
__global__ void finalize_kernel(float* __restrict__ xl, int dout, int ld,
                                const float* __restrict__ s,
                                const float* __restrict__ deg,
                                const int* __restrict__ lab,
                                const float* __restrict__ dW2,
                                const float* __restrict__ db2,
                                const float* __restrict__ bias,
                                const float* __restrict__ gamma,
                                const float* __restrict__ beta,
                                const float* __restrict__ mean,
                                const float* __restrict__ var,
                                int do_bn) {
    int wid  = (blockIdx.x * blockDim.x + threadIdx.x) >> 5;
    int lane = threadIdx.x & 31;
    if (wid >= N_NODES) return;

    int l = lab[wid];
    int c = l < 0 ? 0 : l;
    float inv = 1.0f / deg[wid];
    float mh  = s[(size_t)wid * ENC_H + lane] * inv;
    float* xr = xl + (size_t)wid * ld;
    const float* w2 = dW2 + (size_t)c * ENC_H * dout;

    int ngroup = (dout + 31) >> 5;
    for (int g = 0; g < ngroup; ++g) {
        int j = (g << 5) + lane;
        bool ok = (j < dout);
        float dec = ok ? db2[c * dout + j] : 0.0f;
        for (int h = 0; h < ENC_H; ++h) {
            float mhh = __shfl(mh, h);            // uniform, all lanes participate
            if (ok) dec = fmaf(mhh, w2[h * dout + j], dec);
        }
        if (ok) {
            float o = (l < 0 ? xr[j] : dec) + bias[j];
            if (do_bn) {
                o = (o - mean[j]) * rsqrtf(var[j] + BN_EPS) * gamma[j] + beta[j];
                o = fmaxf(o, 0.0f);
            }
            xr[j] = o;
        }
    }
}

// =====================================================================
// Row softmax over 10 classes (logits stored with stride 16)
// =====================================================================
__global__ void softmax_kernel(const float* __restrict__ logits, float* __restrict__ out) {
    int n = blockIdx.x * blockDim.x + threadIdx.x;
    if (n >= N_NODES) return;
    const float* r = logits + (size_t)n * 16;
    float m = r[0];
    #pragma unroll
    for (int i = 1; i < NCLS; ++i) m = fmaxf(m, r[i]);
    float e[NCLS], sum = 0.0f;
    #pragma unroll
    for (int i = 0; i < NCLS; ++i) { e[i] = expf(r[i] - m); sum += e[i]; }
    float is = 1.0f / sum;
    #pragma unroll
    for (int i = 0; i < NCLS; ++i) out[(size_t)n * NCLS + i] = e[i] * is;
}

// =====================================================================
// host driver
// =====================================================================
extern "C" void kernel_launch(void* const* d_in, const int* in_sizes, int n_in,
                              void* d_out, int out_size, void* d_ws, size_t ws_size,
                              hipStream_t stream) {
    const float* x   = (const float*)d_in[0];
    const int*   ei  = (const int*)d_in[1];
    const int*   lab = (const int*)d_in[2];
    // params[l]: lin_W, bias, eW1, eb1, eW2, eb2, dW1, db1, dW2, db2  (insertion order)
    auto P  = [&](int layer, int k) { return (const float*)d_in[3 + layer * 10 + k]; };
    // bn[b]: gamma, beta, mean, var
    auto BN = [&](int b, int k)     { return (const float*)d_in[3 + 3 * 10 + b * 4 + k]; };

    // ---- workspace carve-up (all 256B aligned) ----
    char* ws = (char*)d_ws;
    size_t off = 0;
    auto carve = [&](size_t bytes) { char* p = ws + off; off += (bytes + 255) & ~(size_t)255; return p; };
    float* buf0 = (float*)carve((size_t)N_NODES * 128 * 4);   // layer0 activations
    float* buf1 = (float*)carve((size_t)N_NODES * 128 * 4);   // layer1 activations
    float* xl2  = (float*)carve((size_t)N_NODES * 16 * 4);    // layer2 logits (pad 16)
    float* sbuf = (float*)carve((size_t)N_NODES * ENC_H * 4); // edge aggregation
    float* enc  = (float*)carve((size_t)N_NODES * ENC_OUT * 4);
    float* deg  = (float*)carve((size_t)N_NODES * 4);
    float* dinv = (float*)carve((size_t)N_NODES * 4);
    (void)ws_size; (void)in_sizes; (void)n_in; (void)out_size;

    const int TB = 256;
    const int nodeWaveBlocks = (N_NODES * 32 + TB - 1) / TB;  // wave-per-node kernels

    // ---- gcn_norm precompute ----
    init_deg_kernel<<<(N_NODES + TB - 1) / TB, TB, 0, stream>>>(deg);
    edge_deg_kernel<<<(N_EDGES + TB - 1) / TB, TB, 0, stream>>>(ei, deg);
    dinv_kernel<<<(N_NODES + TB - 1) / TB, TB, 0, stream>>>(deg, dinv);

    const int MT = N_NODES / 16;     // 6250 M tiles exactly

    struct L { const float* xin; float* xl; int dout; int ld; int bn; };
    L layers[3] = {
        { x,    buf0, HID,  128, 0 },
        { buf0, buf1, HID,  128, 1 },
        { buf1, xl2,  NCLS, 16,  -1 },
    };

    for (int li = 0; li < 3; ++li) {
        const L& Lc = layers[li];
        // zero scatter buffer
        zero_kernel<<<(N_NODES * ENC_H + TB - 1) / TB, TB, 0, stream>>>(sbuf, N_NODES * ENC_H);
        // xl = x @ lin_W  (f32 WMMA)
        gemm_wmma_kernel<<<512, TB, 0, stream>>>(Lc.xin, P(li, 0), Lc.xl, Lc.dout, Lc.ld, MT);
        // per-node routed encoder -> enc[N,4]
        encoder_kernel<<<nodeWaveBlocks, TB, 0, stream>>>(Lc.xl, Lc.dout, Lc.ld, lab,
                                                          P(li, 2), P(li, 3), P(li, 4), P(li, 5), enc);
        // per-edge message + scatter-sum into s
        edge_kernel<<<4096, TB, 0, stream>>>(ei, lab, dinv, enc, P(li, 6), P(li, 7), sbuf);
        // decoder + bias (+ BN + relu), in place on xl
        int do_bn = (Lc.bn >= 0) ? 1 : 0;
        const float* g  = do_bn ? BN(Lc.bn, 0) : (const float*)nullptr;
        const float* b  = do_bn ? BN(Lc.bn, 1) : (const float*)nullptr;
        const float* mu = do_bn ? BN(Lc.bn, 2) : (const float*)nullptr;
        const float* vr = do_bn ? BN(Lc.bn, 3) : (const float*)nullptr;
        finalize_kernel<<<nodeWaveBlocks, TB, 0, stream>>>(Lc.xl, Lc.dout, Lc.ld, sbuf, deg, lab,
                                                           P(li, 8), P(li, 9), P(li, 1),
                                                           g, b, mu, vr, do_bn);
    }

    softmax_kernel<<<(N_NODES + TB - 1) / TB, TB, 0, stream>>>(xl2, (float*)d_out);
}